// RandomFGLClassifier_27376121544990
// MI455X (gfx1250) — compile-verified
//
#include <hip/hip_runtime.h>

#define IN_F   212445
#define N_BATCH 32
#define NCLS   20

// workspace float offsets
#define WS_T2    0        // [32*128]  t2[n*128 + m]
#define WS_CW1   4096     // [128]
#define WS_CNT2  4224     // [128]
#define WS_U2    4352     // [128]
#define WS_V2    4480     // [128]
#define WS_W2    4608     // [128]
#define WS_APAD  4736     // [32*128]  A[q*128 + m], rows 20..31 stay zero
#define WS_CQ    8832     // [20]
#define WS_NZERO 8832     // zero t2, cw1, cnt2, (u2/v2/w2 overwritten), Apad

typedef __attribute__((ext_vector_type(2))) float v2f;
typedef __attribute__((ext_vector_type(8))) float v8f;

// ---------------------------------------------------------------- init
__global__ void init_zero(float* __restrict__ F) {
  int idx = blockIdx.x * 256 + threadIdx.x;
  if (idx < WS_NZERO) F[idx] = 0.0f;
}

// ------------------------------------------------- bias-path histograms
// cw1[m]  = #{ j in [0,65536) : parent2[parent1[j]] == m }
// cnt2[m] = #{ k in [0,16384) : parent2[k] == m }
__global__ __launch_bounds__(256) void hist_kernel(
    const int* __restrict__ p1, const int* __restrict__ p2,
    float* __restrict__ cw1, float* __restrict__ cnt2) {
  __shared__ float h1[128];
  __shared__ float h2[128];
  int tid = threadIdx.x;
  if (tid < 128) { h1[tid] = 0.0f; h2[tid] = 0.0f; }
  __syncthreads();
  int j = blockIdx.x * 256 + tid;
  if (j < 65536) atomicAdd(&h1[p2[p1[j]]], 1.0f);
  if (j < 16384) atomicAdd(&h2[p2[j]], 1.0f);
  __syncthreads();
  if (tid < 128) {
    if (h1[tid] != 0.0f) atomicAdd(&cw1[tid], h1[tid]);
    if (h2[tid] != 0.0f) atomicAdd(&cnt2[tid], h2[tid]);
  }
}

// ---------------------------------------------- main HBM-bound reduction
// t2[n*128+m] = sum_{i : parent2[parent1[parent0[i]]] == m} x[n, i]
// Each thread owns one i per iteration (coalesced across lanes), chases the
// 3-level parent map once (L2-resident tables), then loops batch n.
#define CHUNK_ITERS 8
#define CHUNK (256 * CHUNK_ITERS)

__global__ __launch_bounds__(256) void fgl_reduce(
    const float* __restrict__ x,
    const int* __restrict__ p0, const int* __restrict__ p1,
    const int* __restrict__ p2, float* __restrict__ t2) {
  __shared__ float bins[N_BATCH * 128];   // bins[n*128 + m], 16 KB
  int tid = threadIdx.x;
  for (int s = tid; s < N_BATCH * 128; s += 256) bins[s] = 0.0f;
  __syncthreads();

  int base = blockIdx.x * CHUNK;
  for (int it = 0; it < CHUNK_ITERS; ++it) {
    int i = base + it * 256 + tid;
    if (i < IN_F) {
      int m = p2[p1[p0[i]]];
      const float* xp = x + i;
#pragma unroll 8
      for (int n = 0; n < N_BATCH; ++n) {
        atomicAdd(&bins[n * 128 + m], xp[(size_t)n * IN_F]);
      }
    }
  }
  __syncthreads();
  for (int s = tid; s < N_BATCH * 128; s += 256) {
    float v = bins[s];
    if (v != 0.0f) atomicAdd(&t2[s], v);
  }
}

// ------------------------------------------ fold weight chain (tiny)
// u1 = W1@W0, v1 = W1@b0 ; u2 = W2@u1, v2 = W2@v1, w2 = W2@b1
__global__ __launch_bounds__(128) void small_vecs(
    const float* __restrict__ W0, const float* __restrict__ b0,
    const float* __restrict__ W1, const float* __restrict__ b1,
    const float* __restrict__ W2,
    float* __restrict__ u2, float* __restrict__ v2, float* __restrict__ w2) {
  __shared__ float u1[64];
  __shared__ float v1[64];
  int t = threadIdx.x;
  if (t < 64) {
    float a = 0.0f, b = 0.0f;
    for (int c = 0; c < 32; ++c) {
      float w = W1[t * 32 + c];
      a += w * W0[c];
      b += w * b0[c];
    }
    u1[t] = a; v1[t] = b;
  }
  __syncthreads();
  float a = 0.0f, b = 0.0f, c2 = 0.0f;
  for (int c = 0; c < 64; ++c) {
    float w = W2[t * 64 + c];
    a  += w * u1[c];
    b  += w * v1[c];
    c2 += w * b1[c];
  }
  u2[t] = a; v2[t] = b; w2[t] = c2;
}

// ------------------------- fold fc_w against u2/v2/w2/b2  (1.3 MB read)
// Apad[q,m] = sum_o fc_w[q, o*128+m] * u2[o]            (q < 20)
// const_q   = sum_m ( cw1[m]*sum_o fc_w*v2 + cnt2[m]*sum_o fc_w*w2
//                     + sum_o fc_w*b2 ) + fc_b[q]
__global__ __launch_bounds__(128) void afc(
    const float* __restrict__ fc_w, const float* __restrict__ fc_b,
    const float* __restrict__ b2,
    const float* __restrict__ u2, const float* __restrict__ v2,
    const float* __restrict__ w2,
    const float* __restrict__ cw1, const float* __restrict__ cnt2,
    float* __restrict__ Apad, float* __restrict__ constq) {
  __shared__ float red[128];
  int q = blockIdx.x;
  int m = threadIdx.x;
  const float* fw = fc_w + (size_t)q * 16384 + m;
  float aA = 0.0f, aV = 0.0f, aW = 0.0f, aB = 0.0f;
#pragma unroll 4
  for (int o = 0; o < 128; ++o) {
    float w = fw[o * 128];   // coalesced over m
    aA += w * u2[o];
    aV += w * v2[o];
    aW += w * w2[o];
    aB += w * b2[o];
  }
  Apad[q * 128 + m] = aA;
  red[m] = aV * cw1[m] + aW * cnt2[m] + aB;
  __syncthreads();
  for (int s = 64; s > 0; s >>= 1) {
    if (m < s) red[m] += red[m + s];
    __syncthreads();
  }
  if (m == 0) constq[q] = red[0] + fc_b[q];
}

// -------------------------------- final GEMM via V_WMMA_F32_16X16X4_F32
// D[n,q] = sum_m T[n,m] * A'[m,q] ; T[n,m]=t2[n*128+m], A'[m,q]=Apad[q*128+m]
// M=32 (2 tiles), N=32 (2 tiles, rows q>=20 of Apad are zero), K=128.
// All fragment loads unconditional -> no exec-mask churn in the inner loop.
__global__ __launch_bounds__(32) void final_gemm(
    const float* __restrict__ t2, const float* __restrict__ Apad,
    const float* __restrict__ constq, float* __restrict__ out) {
  int l    = threadIdx.x;      // lane 0..31
  int half = l >> 4;           // K-half per ISA 16x4 A/B layout
  int lm   = l & 15;

  v8f acc00 = {}, acc01 = {}, acc10 = {}, acc11 = {};

  const float* tA0 = t2 + lm * 128;          // rows n = lm
  const float* tA1 = t2 + (16 + lm) * 128;   // rows n = 16+lm
  const float* pB0 = Apad + lm * 128;        // cols q = lm
  const float* pB1 = Apad + (16 + lm) * 128; // cols q = 16+lm (zero if >=20)

  for (int kk = 0; kk < 32; ++kk) {
    int k = kk * 4 + 2 * half;               // even -> 8B-aligned pair loads
    v2f a0, a1, bv0, bv1;
    a0.x  = tA0[k];  a0.y  = tA0[k + 1];
    a1.x  = tA1[k];  a1.y  = tA1[k + 1];
    bv0.x = pB0[k];  bv0.y = pB0[k + 1];
    bv1.x = pB1[k];  bv1.y = pB1[k + 1];

    acc00 = __builtin_amdgcn_wmma_f32_16x16x4_f32(false, a0, false, bv0,
                                                  (short)0, acc00, false, false);
    acc01 = __builtin_amdgcn_wmma_f32_16x16x4_f32(false, a0, false, bv1,
                                                  (short)0, acc01, false, false);
    acc10 = __builtin_amdgcn_wmma_f32_16x16x4_f32(false, a1, false, bv0,
                                                  (short)0, acc10, false, false);
    acc11 = __builtin_amdgcn_wmma_f32_16x16x4_f32(false, a1, false, bv1,
                                                  (short)0, acc11, false, false);
  }

  // C/D layout: VGPR r -> lanes0-15: M=r, N=lane ; lanes16-31: M=r+8, N=lane-16
#pragma unroll
  for (int r = 0; r < 8; ++r) {
    int M  = r + 8 * half;
    int q0 = lm;
    int q1 = 16 + lm;
    int n0 = M;
    int n1 = 16 + M;
    out[n0 * NCLS + q0] = acc00[r] + constq[q0];
    out[n1 * NCLS + q0] = acc10[r] + constq[q0];
    if (q1 < NCLS) {
      out[n0 * NCLS + q1] = acc01[r] + constq[q1];
      out[n1 * NCLS + q1] = acc11[r] + constq[q1];
    }
  }
}

// ---------------------------------------------------------------- launch
extern "C" void kernel_launch(void* const* d_in, const int* in_sizes, int n_in,
                              void* d_out, int out_size, void* d_ws, size_t ws_size,
                              hipStream_t stream) {
  const float* x    = (const float*)d_in[0];
  const float* W0   = (const float*)d_in[1];
  const float* b0   = (const float*)d_in[2];
  const float* W1   = (const float*)d_in[3];
  const float* b1   = (const float*)d_in[4];
  const float* W2   = (const float*)d_in[5];
  const float* b2   = (const float*)d_in[6];
  const float* fc_w = (const float*)d_in[7];
  const float* fc_b = (const float*)d_in[8];
  const int*   p0   = (const int*)d_in[9];
  const int*   p1   = (const int*)d_in[10];
  const int*   p2   = (const int*)d_in[11];
  float* F   = (float*)d_ws;
  float* out = (float*)d_out;

  init_zero<<<dim3((WS_NZERO + 255) / 256), dim3(256), 0, stream>>>(F);
  hist_kernel<<<dim3(65536 / 256), dim3(256), 0, stream>>>(
      p1, p2, F + WS_CW1, F + WS_CNT2);
  fgl_reduce<<<dim3((IN_F + CHUNK - 1) / CHUNK), dim3(256), 0, stream>>>(
      x, p0, p1, p2, F + WS_T2);
  small_vecs<<<dim3(1), dim3(128), 0, stream>>>(
      W0, b0, W1, b1, W2, F + WS_U2, F + WS_V2, F + WS_W2);
  afc<<<dim3(NCLS), dim3(128), 0, stream>>>(
      fc_w, fc_b, b2, F + WS_U2, F + WS_V2, F + WS_W2,
      F + WS_CW1, F + WS_CNT2, F + WS_APAD, F + WS_CQ);
  final_gemm<<<dim3(1), dim3(32), 0, stream>>>(
      F + WS_T2, F + WS_APAD, F + WS_CQ, out);
}